// SelfAttentionANN16_16466904613583
// MI455X (gfx1250) — compile-verified
//
#include <hip/hip_runtime.h>
#include <hip/hip_bf16.h>
#include <math.h>

// ---------------------------------------------------------------------------
// Fused SelfAttention(16) -> Linear(16,64)+sigmoid -> Linear(64,8)
//       -> SelfAttention(8) -> softmax -> dot(x[8:16])
// CDNA5 (gfx1250), wave32. Dense layers run on V_WMMA_F32_16X16X4_F32.
// ---------------------------------------------------------------------------

typedef float v2f __attribute__((ext_vector_type(2)));
typedef float v8f __attribute__((ext_vector_type(8)));

#define WAVES_PER_BLOCK 8
#define ROWS_PER_WAVE 32
#define ROWS_PER_BLOCK (WAVES_PER_BLOCK * ROWS_PER_WAVE) // 256

__device__ __forceinline__ float fast_rcp(float x) {
  return __builtin_amdgcn_rcpf(x);
}

__global__ __launch_bounds__(256) void fused_attn_mlp_kernel(
    const float* __restrict__ x,   // [B,16]
    const float* __restrict__ wq,  // [2]
    const float* __restrict__ wk,  // [2]
    const float* __restrict__ wv,  // [2]
    const float* __restrict__ Wh,  // [64,16]
    const float* __restrict__ bh,  // [64]
    const float* __restrict__ Wo,  // [8,64]
    const float* __restrict__ bo,  // [8]
    float* __restrict__ out,       // [B]
    int B) {
  // Per-wave LDS staging (wave-private slabs; barriers are belt+braces for the
  // compiler, HW keeps same-wave DS ops in order).
  __shared__ __attribute__((aligned(16))) float lds_o1[WAVES_PER_BLOCK][32][18]; // stride 18 (even -> b64 aligned)
  __shared__ __attribute__((aligned(16))) float lds_x1[WAVES_PER_BLOCK][16][66]; // stride 66
  __shared__ __attribute__((aligned(16))) float lds_x2[WAVES_PER_BLOCK][32][9];

  const int lane = threadIdx.x & 31;
  const int wave = threadIdx.x >> 5;
  const int h    = lane >> 4;   // lane half: 0 or 1
  const int nl   = lane & 15;   // lane % 16

  const int row    = blockIdx.x * ROWS_PER_BLOCK + wave * ROWS_PER_WAVE + lane;
  const int rclamp = (row < B) ? row : (B - 1);

  // Prefetch this lane's row while we chew on the weight preamble.
  __builtin_prefetch(x + (size_t)rclamp * 16, 0, 3);

  // ---- scalar attention weights --------------------------------------------
  const float q0 = wq[0], q1 = wq[1];
  const float k0 = wk[0], k1 = wk[1];
  const float v0 = wv[0], v1 = wv[1];
  const float c1 = q0 * k0;  // Alpha1[i,j] = c1 * x_i * x_j
  const float c2 = q1 * k1;

  // ---- preload GEMM weights into WMMA B-layout registers -------------------
  // B-matrix 4x16 layout: VGPR v, lane l -> K = v + 2*(l/16), N = l%16.
  // GEMM1: B1[k][n] = Wh[n][k]   (Wh is [64,16] row-major), N split in 4 tiles.
  v2f wb1[4][4];  // [k-chunk][n-tile]
#pragma unroll
  for (int kc = 0; kc < 4; ++kc) {
#pragma unroll
    for (int nt = 0; nt < 4; ++nt) {
      const int n = nt * 16 + nl;
      const int k = kc * 4 + 2 * h;           // even -> 8B aligned pair
      wb1[kc][nt] = *(const v2f*)(Wh + n * 16 + k);
    }
  }
  // GEMM2: B2[k][n] = Wo[n][k]  (Wo is [8,64]); pad N 8->16 with zeros.
  v2f wb2[16];
#pragma unroll
  for (int kc = 0; kc < 16; ++kc) {
    const int k = kc * 4 + 2 * h;
    if (nl < 8) {
      wb2[kc] = *(const v2f*)(Wo + nl * 64 + k);
    } else {
      wb2[kc] = (v2f){0.0f, 0.0f};
    }
  }
  // Bias values in C/D layout (per lane: column n = nl (+16*nt)).
  float bh4[4];
#pragma unroll
  for (int nt = 0; nt < 4; ++nt) bh4[nt] = bh[nt * 16 + nl];
  const float bo1 = (nl < 8) ? bo[nl] : 0.0f;

  // ---- load this lane's input row (coalesced: 64B/lane, 4x b128) -----------
  float xv[16];
  {
    const float4* xp = (const float4*)(x + (size_t)rclamp * 16);
#pragma unroll
    for (int i = 0; i < 4; ++i) {
      float4 t = xp[i];
      xv[4 * i + 0] = t.x; xv[4 * i + 1] = t.y;
      xv[4 * i + 2] = t.z; xv[4 * i + 3] = t.w;
    }
  }

  // ========================= Attention block 1 (D=16) ========================
  // A[i,j] = softmax_j(c1*x_i*x_j);  o1[j] = sum_i (v0*x_i) * A[i,j]
  float xmax = xv[0], xmin = xv[0];
#pragma unroll
  for (int j = 1; j < 16; ++j) {
    xmax = fmaxf(xmax, xv[j]);
    xmin = fminf(xmin, xv[j]);
  }
  float o1[16];
#pragma unroll
  for (int j = 0; j < 16; ++j) o1[j] = 0.0f;
#pragma unroll
  for (int i = 0; i < 16; ++i) {
    const float s = c1 * xv[i];
    const float m = (s > 0.0f) ? s * xmax : s * xmin;  // row max for stability
    float e[16];
    float Z = 0.0f;
#pragma unroll
    for (int j = 0; j < 16; ++j) {
      e[j] = __expf(fmaf(s, xv[j], -m));
      Z += e[j];
    }
    const float g = v0 * xv[i] * fast_rcp(Z);
#pragma unroll
    for (int j = 0; j < 16; ++j) o1[j] = fmaf(g, e[j], o1[j]);
  }

  // Stage o1 (32 rows x 16) for WMMA A-layout reads.
#pragma unroll
  for (int j = 0; j < 16; ++j) lds_o1[wave][lane][j] = o1[j];
  __syncthreads();

  // ======================== Dense layers via WMMA ============================
  // Two M=16 row tiles per wave.
#pragma unroll
  for (int mt = 0; mt < 2; ++mt) {
    // ---- GEMM1: [16x16] x [16x64] + bh, then sigmoid ----
    v8f acc[4];
#pragma unroll
    for (int nt = 0; nt < 4; ++nt) {
#pragma unroll
      for (int v = 0; v < 8; ++v) acc[nt][v] = bh4[nt];
    }
#pragma unroll
    for (int kc = 0; kc < 4; ++kc) {
      // A-layout 16x4: VGPR v, lane l -> M=l%16, K = v + 2*(l/16)
      const v2f a = *(const v2f*)&lds_o1[wave][mt * 16 + nl][kc * 4 + 2 * h];
#pragma unroll
      for (int nt = 0; nt < 4; ++nt) {
        acc[nt] = __builtin_amdgcn_wmma_f32_16x16x4_f32(
            /*neg_a=*/false, a, /*neg_b=*/false, wb1[kc][nt],
            /*c_mod=*/(short)0, acc[nt], /*reuse_a=*/false, /*reuse_b=*/false);
      }
    }
    __syncthreads();  // lds_x1 slab reuse between mt iterations
    // Sigmoid in D-layout and restage as row-major x1 tile [16 x 64].
#pragma unroll
    for (int nt = 0; nt < 4; ++nt) {
#pragma unroll
      for (int v = 0; v < 8; ++v) {
        const float val = fast_rcp(1.0f + __expf(-acc[nt][v]));
        lds_x1[wave][v + 8 * h][nt * 16 + nl] = val;  // row = v+8h, col = 16nt+nl
      }
    }
    __syncthreads();

    // ---- GEMM2: [16x64] x [64x8(pad16)] + bo ----
    v8f acc2;
#pragma unroll
    for (int v = 0; v < 8; ++v) acc2[v] = bo1;
#pragma unroll
    for (int kc = 0; kc < 16; ++kc) {
      const v2f a2 = *(const v2f*)&lds_x1[wave][nl][kc * 4 + 2 * h];
      acc2 = __builtin_amdgcn_wmma_f32_16x16x4_f32(
          false, a2, false, wb2[kc], (short)0, acc2, false, false);
    }
    // Store x2 tile (only columns 0..7 are real).
    if (nl < 8) {
#pragma unroll
      for (int v = 0; v < 8; ++v)
        lds_x2[wave][mt * 16 + v + 8 * h][nl] = acc2[v];
    }
  }
  __syncthreads();

  // ========================= Attention block 2 (D=8) =========================
  float y[8];
#pragma unroll
  for (int j = 0; j < 8; ++j) y[j] = lds_x2[wave][lane][j];
  float ymax = y[0], ymin = y[0];
#pragma unroll
  for (int j = 1; j < 8; ++j) {
    ymax = fmaxf(ymax, y[j]);
    ymin = fminf(ymin, y[j]);
  }
  float o2[8];
#pragma unroll
  for (int j = 0; j < 8; ++j) o2[j] = 0.0f;
#pragma unroll
  for (int i = 0; i < 8; ++i) {
    const float s = c2 * y[i];
    const float m = (s > 0.0f) ? s * ymax : s * ymin;
    float e[8];
    float Z = 0.0f;
#pragma unroll
    for (int j = 0; j < 8; ++j) {
      e[j] = __expf(fmaf(s, y[j], -m));
      Z += e[j];
    }
    const float g = v1 * y[i] * fast_rcp(Z);
#pragma unroll
    for (int j = 0; j < 8; ++j) o2[j] = fmaf(g, e[j], o2[j]);
  }

  // Final: softmax(o2) dotted with x[8:16]  ->  (sum e_j * xhi_j) / (sum e_j)
  float m2 = o2[0];
#pragma unroll
  for (int j = 1; j < 8; ++j) m2 = fmaxf(m2, o2[j]);
  float Z2 = 0.0f, num = 0.0f;
#pragma unroll
  for (int j = 0; j < 8; ++j) {
    const float e = __expf(o2[j] - m2);
    Z2 += e;
    num = fmaf(e, xv[8 + j], num);
  }
  const float result = num * fast_rcp(Z2);

  if (row < B) out[row] = result;
}

extern "C" void kernel_launch(void* const* d_in, const int* in_sizes, int n_in,
                              void* d_out, int out_size, void* d_ws, size_t ws_size,
                              hipStream_t stream) {
  const float* x  = (const float*)d_in[0];
  const float* wq = (const float*)d_in[1];
  const float* wk = (const float*)d_in[2];
  const float* wv = (const float*)d_in[3];
  const float* Wh = (const float*)d_in[4];
  const float* bh = (const float*)d_in[5];
  const float* Wo = (const float*)d_in[6];
  const float* bo = (const float*)d_in[7];
  float* out = (float*)d_out;

  const int B = in_sizes[0] / 16;  // 524288
  const int grid = (B + ROWS_PER_BLOCK - 1) / ROWS_PER_BLOCK;  // 2048

  fused_attn_mlp_kernel<<<grid, 256, 0, stream>>>(x, wq, wk, wv, Wh, bh, Wo, bo,
                                                  out, B);
}